// SPDBatchNormMean_22643067584713
// MI455X (gfx1250) — compile-verified
//
#include <hip/hip_runtime.h>
#include <math.h>

// ---------------------------------------------------------------------------
// SPD BatchNorm (Karcher mean, n_iter=1) for MI455X / gfx1250.
// B = 8192 SPD matrices, n = 64, f32.  All 64x64 matmuls run on
// V_WMMA_F32_16X16X4_F32; x tiles stream through the CDNA5 async
// global->LDS path (ASYNCcnt) with double buffering.
// ---------------------------------------------------------------------------

typedef __attribute__((ext_vector_type(2))) float v2f;
typedef __attribute__((ext_vector_type(8))) float v8f;

#define NTHR 128          // 4 waves of 32
#define LDAP 68           // LDS row stride in floats (16B-aligned rows)
#define MATB (64 * LDAP)  // floats per LDS matrix buffer

// workspace layout (float offsets)
#define OFF_SUM  0        // batch sum of x  (4096)
#define OFF_GS   4096     // mean^{1/2}
#define OFF_GIS  8192     // mean^{-1/2}
#define OFF_MSUM 12288    // sum of logm(S_i)
#define OFF_W    16384    // W = bias^{1/2} @ mean_new^{-1/2}
#define WS_FLOATS 20480

// ---------------------------------------------------------------------------
// LDS <-> global 64x64 moves (synchronous, 128-bit)
// ---------------------------------------------------------------------------
__device__ inline void ld_mat(float* dst, const float* src) {
  const float4* s4 = (const float4*)src;
  for (int f = threadIdx.x; f < 1024; f += NTHR) {
    float4 v = s4[f];
    int row = f >> 4, col = (f & 15) << 2;
    *(float4*)(dst + row * LDAP + col) = v;
  }
}
__device__ inline void st_mat(const float* src, float* dst) {
  float4* d4 = (float4*)dst;
  for (int f = threadIdx.x; f < 1024; f += NTHR) {
    int row = f >> 4, col = (f & 15) << 2;
    d4[f] = *(const float4*)(src + row * LDAP + col);
  }
}

// ---------------------------------------------------------------------------
// Async global -> LDS (CDNA5, tracked by ASYNCcnt).  Each wave issues exactly
// 8 b128 ops per 16KB tile; async loads complete in order, so
// s_wait_asynccnt<=8 means "previous tile landed" while the next is in flight.
// ---------------------------------------------------------------------------
__device__ inline void ld_mat_async(float* dst, const float* src) {
  for (int f = threadIdx.x; f < 1024; f += NTHR) {
    int row = f >> 4, col = (f & 15) << 2;
    unsigned lds = (unsigned)(size_t)(dst + row * LDAP + col);
    const float* g = src + 4 * (size_t)f;
    asm volatile("global_load_async_to_lds_b128 %0, %1, off"
                 :: "v"(lds), "v"(g) : "memory");
  }
}
__device__ inline void async_wait_prev() {   // <=8 outstanding (next tile)
  asm volatile("s_wait_asynccnt 8" ::: "memory");
}
__device__ inline void async_wait_all() {
  asm volatile("s_wait_asynccnt 0" ::: "memory");
}

// ---------------------------------------------------------------------------
// 64x64x64 f32 GEMM, one workgroup = 4 waves; wave w owns C row-block w.
// A fragments hoisted once (b64 LDS loads); B fragments are software-
// pipelined by the scheduler ahead of each WMMA.  (The 4-concurrent-tile
// variant spilled addresses to scratch; this form stays spill-free.)
// ---------------------------------------------------------------------------
template <bool BT>
__device__ inline void gemm64(const float* A, const float* B, float* C) {
  __syncthreads();
  const int wave = threadIdx.x >> 5;
  const int lane = threadIdx.x & 31;
  const int half = lane >> 4;
  const int idx  = lane & 15;
  const int mrow = wave * 16 + idx;
  v2f af[16];
  #pragma unroll
  for (int ks = 0; ks < 16; ++ks)
    af[ks] = *(const v2f*)(A + mrow * LDAP + 4 * ks + 2 * half);
  #pragma unroll 1
  for (int nt = 0; nt < 4; ++nt) {
    const int ncol = nt * 16 + idx;
    v2f bf[16];
    #pragma unroll
    for (int ks = 0; ks < 16; ++ks) {
      const int kk = 4 * ks + 2 * half;
      if (BT) {
        bf[ks] = *(const v2f*)(B + ncol * LDAP + kk);
      } else {
        bf[ks][0] = B[kk * LDAP + ncol];
        bf[ks][1] = B[(kk + 1) * LDAP + ncol];
      }
    }
    v8f acc = {};
    #pragma unroll
    for (int ks = 0; ks < 16; ++ks)
      acc = __builtin_amdgcn_wmma_f32_16x16x4_f32(
          false, af[ks], false, bf[ks], (short)0, acc, false, false);
    #pragma unroll
    for (int r = 0; r < 8; ++r)
      C[(wave * 16 + half * 8 + r) * LDAP + ncol] = acc[r];
  }
  __syncthreads();
}

// ---------------------------------------------------------------------------
// helpers (all threads uniform)
// ---------------------------------------------------------------------------
__device__ inline float trace64(const float* A, float* red) {
  __syncthreads();
  if (threadIdx.x < 64) red[threadIdx.x] = A[threadIdx.x * LDAP + threadIdx.x];
  __syncthreads();
  float s = 0.f;
  #pragma unroll
  for (int i = 0; i < 64; ++i) s += red[i];
  return s;
}
__device__ inline void ew_identity(float* A) {
  for (int e = threadIdx.x; e < 4096; e += NTHR) {
    int r = e >> 6, c = e & 63;
    A[r * LDAP + c] = (r == c) ? 1.f : 0.f;
  }
}
__device__ inline void ew_scale(float* A, float s) {
  for (int e = threadIdx.x; e < 4096; e += NTHR) {
    int r = e >> 6, c = e & 63;
    A[r * LDAP + c] *= s;
  }
}
__device__ inline void ew_copy_scale(float* D, const float* S, float s) {
  for (int e = threadIdx.x; e < 4096; e += NTHR) {
    int r = e >> 6, c = e & 63;
    D[r * LDAP + c] = S[r * LDAP + c] * s;
  }
}

// Coupled Newton-Schulz: input A in Y (eigs in (0,3)); out Y~A^{1/2}, Z~A^{-1/2}
__device__ inline void ns_sqrt(float* Y, float* Z, float* T, float* U, float* V,
                               int iters) {
  ew_identity(Z);
  float *y = Y, *z = Z, *u = U, *v = V;
  for (int it = 0; it < iters; ++it) {
    gemm64<false>(z, y, T);
    for (int e = threadIdx.x; e < 4096; e += NTHR) {  // T = 1.5I - 0.5T
      int r = e >> 6, c = e & 63;
      float tv = T[r * LDAP + c];
      T[r * LDAP + c] = (r == c ? 1.5f : 0.f) - 0.5f * tv;
    }
    gemm64<false>(y, T, u);
    gemm64<false>(T, z, v);
    float* p;
    p = y; y = u; u = p;
    p = z; z = v; v = p;
  }
  if (y != Y) {
    __syncthreads();
    for (int e = threadIdx.x; e < 4096; e += NTHR) {
      int r = e >> 6, c = e & 63;
      Y[r * LDAP + c] = y[r * LDAP + c];
      Z[r * LDAP + c] = z[r * LDAP + c];
    }
    __syncthreads();
  }
}

// ---------------------------------------------------------------------------
__global__ void kzero(float* p, int n) {
  int i = blockIdx.x * blockDim.x + threadIdx.x;
  if (i < n) p[i] = 0.f;
}

// Kernel 1: batch sum of x into ws[OFF_SUM] (b128 streaming loads)
__global__ void kmean_sum(const float* __restrict__ x, float* ws, int Bn) {
  float4 acc[8];
  #pragma unroll
  for (int j = 0; j < 8; ++j) acc[j] = make_float4(0.f, 0.f, 0.f, 0.f);
  const int t = threadIdx.x;
  for (int m = blockIdx.x; m < Bn; m += gridDim.x) {
    const float4* p4 = (const float4*)(x + (size_t)m * 4096);
    #pragma unroll
    for (int j = 0; j < 8; ++j) {
      float4 v = p4[t + 128 * j];
      acc[j].x += v.x; acc[j].y += v.y; acc[j].z += v.z; acc[j].w += v.w;
    }
  }
  #pragma unroll
  for (int j = 0; j < 8; ++j) {
    int e = 4 * (t + 128 * j);
    atomicAdd(&ws[OFF_SUM + e + 0], acc[j].x);
    atomicAdd(&ws[OFF_SUM + e + 1], acc[j].y);
    atomicAdd(&ws[OFF_SUM + e + 2], acc[j].z);
    atomicAdd(&ws[OFF_SUM + e + 3], acc[j].w);
  }
}

// Kernel 2 (1 block): Gs = mean^{1/2}, Gis = mean^{-1/2} via Newton-Schulz
__global__ void kprep1(float* ws, int Bn) {
  __shared__ __align__(16) float M[MATB], Y[MATB], Z[MATB], T[MATB], U[MATB], V[MATB];
  __shared__ float red[64];
  const float invB = 1.f / (float)Bn;
  for (int e = threadIdx.x; e < 4096; e += NTHR) {
    int r = e >> 6, c = e & 63;
    M[r * LDAP + c] = ws[OFF_SUM + e] * invB;
  }
  float tr = trace64(M, red);
  float cn = tr * (1.f / 48.f);
  ew_copy_scale(Y, M, 1.f / cn);
  ns_sqrt(Y, Z, T, U, V, 14);
  const float sc = sqrtf(cn), isc = rsqrtf(cn);
  for (int e = threadIdx.x; e < 4096; e += NTHR) {
    int r = e >> 6, c = e & 63;
    ws[OFF_GS  + e] = Y[r * LDAP + c] * sc;
    ws[OFF_GIS + e] = Z[r * LDAP + c] * isc;
  }
}

// ---------------------------------------------------------------------------
// Kernel 3: sum_i logm(Gis x_i Gis); async double-buffered x tiles.
// logm via 3 NS square roots + 4-term Gregory series.
// ---------------------------------------------------------------------------
__global__ void klog(const float* __restrict__ x, float* ws, int Bn) {
  __shared__ __align__(16) float GIS[MATB], SUM[MATB];
  __shared__ __align__(16) float X0[MATB], X1[MATB];
  __shared__ __align__(16) float B2[MATB], B3[MATB], B4[MATB], B5[MATB], B6[MATB];
  __shared__ float red[64];
  ld_mat(GIS, ws + OFF_GIS);
  for (int e = threadIdx.x; e < 4096; e += NTHR) {
    int r = e >> 6, c = e & 63;
    SUM[r * LDAP + c] = 0.f;
  }
  int par = 0;
  if (blockIdx.x < (unsigned)Bn)
    ld_mat_async(X0, x + (size_t)blockIdx.x * 4096);
  __syncthreads();
  for (int m = blockIdx.x; m < Bn; m += gridDim.x) {
    float* cur = par ? X1 : X0;
    float* nxt = par ? X0 : X1;
    const int mn = m + gridDim.x;
    if (mn < Bn) {
      ld_mat_async(nxt, x + (size_t)mn * 4096);
      async_wait_prev();
    } else {
      async_wait_all();
    }
    __syncthreads();                     // cur tile landed everywhere
    gemm64<false>(GIS, cur, B2);         // Gis * x
    gemm64<false>(B2, GIS, B3);          // S = Gis * x * Gis   (cur now free)
    float tr = trace64(B3, red);
    float cn = tr * (1.f / 48.f);
    float ls = logf(cn);
    ew_scale(B3, 1.f / cn);              // A, eigs near 1
    #pragma unroll 1
    for (int s = 0; s < 3; ++s)          // A <- A^{1/8}
      ns_sqrt(B3, B4, B5, B6, cur, 5);   // cur reused as scratch
    __syncthreads();
    for (int e = threadIdx.x; e < 4096; e += NTHR) {  // E = A - I
      int r = e >> 6, c = e & 63;
      B4[r * LDAP + c] = B3[r * LDAP + c] - (r == c ? 1.f : 0.f);
    }
    gemm64<false>(B4, B4, B5);           // E^2
    gemm64<false>(B5, B4, B6);           // E^3
    gemm64<false>(B5, B5, cur);          // E^4
    for (int e = threadIdx.x; e < 4096; e += NTHR) {
      int r = e >> 6, c = e & 63;
      int i = r * LDAP + c;
      float l0 = B4[i] - 0.5f * B5[i] + (1.f / 3.f) * B6[i] - 0.25f * cur[i];
      SUM[i] += 8.f * l0 + (r == c ? ls : 0.f);
    }
    __syncthreads();
    par ^= 1;
  }
  for (int e = threadIdx.x; e < 4096; e += NTHR) {
    int r = e >> 6, c = e & 63;
    atomicAdd(&ws[OFF_MSUM + e], SUM[r * LDAP + c]);
  }
}

// ---------------------------------------------------------------------------
// Kernel 4 (1 block): mean2 = Gs expm(M) Gs; Gis2 = mean2^{-1/2};
//                     Bs = bias^{1/2}; W = Bs @ Gis2
// ---------------------------------------------------------------------------
__global__ void kprep2(const float* __restrict__ bias, float* ws, int Bn) {
  __shared__ __align__(16) float A0[MATB], A1[MATB], A2[MATB], A3[MATB];
  __shared__ __align__(16) float A4[MATB], A5[MATB], A6[MATB], A7[MATB];
  __shared__ float red[64];
  const float invB = 1.f / (float)Bn;
  for (int e = threadIdx.x; e < 4096; e += NTHR) {  // E = (Msum/B)/16
    int r = e >> 6, c = e & 63;
    A0[r * LDAP + c] = ws[OFF_MSUM + e] * (invB * (1.f / 16.f));
  }
  gemm64<false>(A0, A0, A1);   // E^2
  gemm64<false>(A1, A0, A2);   // E^3
  gemm64<false>(A1, A1, A3);   // E^4
  for (int e = threadIdx.x; e < 4096; e += NTHR) {  // X = I+E+E2/2+E3/6+E4/24
    int r = e >> 6, c = e & 63;
    int i = r * LDAP + c;
    A4[i] = (r == c ? 1.f : 0.f) + A0[i] + 0.5f * A1[i] +
            (1.f / 6.f) * A2[i] + (1.f / 24.f) * A3[i];
  }
  gemm64<false>(A4, A4, A5);   // square 4x -> expm(M) in A4
  gemm64<false>(A5, A5, A4);
  gemm64<false>(A4, A4, A5);
  gemm64<false>(A5, A5, A4);
  ld_mat(A0, ws + OFF_GS);     // mean2 = Gs expM Gs
  gemm64<false>(A0, A4, A1);
  gemm64<false>(A1, A0, A2);
  for (int e = threadIdx.x; e < 4096; e += NTHR) {  // symmetrize -> A3
    int r = e >> 6, c = e & 63;
    A3[r * LDAP + c] = 0.5f * (A2[r * LDAP + c] + A2[c * LDAP + r]);
  }
  float tr2 = trace64(A3, red);
  float c2 = tr2 * (1.f / 48.f);
  ew_copy_scale(A4, A3, 1.f / c2);
  ns_sqrt(A4, A5, A6, A7, A1, 14);         // A5 = (mean2/c2)^{-1/2}
  __syncthreads();
  ew_scale(A5, rsqrtf(c2));                // Gis2
  ld_mat(A0, bias);                        // Bs = bias^{1/2}
  float tr3 = trace64(A0, red);
  float c3 = tr3 * (1.f / 48.f);
  ew_copy_scale(A1, A0, 1.f / c3);
  ns_sqrt(A1, A2, A3, A6, A7, 14);
  __syncthreads();
  ew_scale(A1, sqrtf(c3));                 // Bs
  gemm64<false>(A1, A5, A2);               // W = Bs @ Gis2
  st_mat(A2, ws + OFF_W);
}

// ---------------------------------------------------------------------------
// Kernel 5: out_i = W x_i W^T  (async double-buffered streaming pass)
// ---------------------------------------------------------------------------
__global__ void kapply(const float* __restrict__ x, const float* __restrict__ ws,
                       float* __restrict__ out, int Bn) {
  __shared__ __align__(16) float WB[MATB], X0[MATB], X1[MATB], T0[MATB], T1[MATB];
  ld_mat(WB, ws + OFF_W);
  int par = 0;
  if (blockIdx.x < (unsigned)Bn)
    ld_mat_async(X0, x + (size_t)blockIdx.x * 4096);
  __syncthreads();
  for (int m = blockIdx.x; m < Bn; m += gridDim.x) {
    float* cur = par ? X1 : X0;
    float* nxt = par ? X0 : X1;
    const int mn = m + gridDim.x;
    if (mn < Bn) {
      ld_mat_async(nxt, x + (size_t)mn * 4096);
      async_wait_prev();
    } else {
      async_wait_all();
    }
    __syncthreads();
    gemm64<false>(WB, cur, T0);   // W * x
    gemm64<true>(T0, WB, T1);     // (W*x) * W^T
    st_mat(T1, out + (size_t)m * 4096);
    __syncthreads();
    par ^= 1;
  }
}

// ---------------------------------------------------------------------------
extern "C" void kernel_launch(void* const* d_in, const int* in_sizes, int n_in,
                              void* d_out, int out_size, void* d_ws, size_t ws_size,
                              hipStream_t stream) {
  const float* x    = (const float*)d_in[0];
  const float* bias = (const float*)d_in[1];
  float* out = (float*)d_out;
  float* ws  = (float*)d_ws;
  const int Bn = in_sizes[0] / 4096;   // 8192

  kzero<<<(WS_FLOATS + 255) / 256, 256, 0, stream>>>(ws, WS_FLOATS);
  kmean_sum<<<256, NTHR, 0, stream>>>(x, ws, Bn);
  kprep1<<<1, NTHR, 0, stream>>>(ws, Bn);
  klog<<<512, NTHR, 0, stream>>>(x, ws, Bn);
  kprep2<<<1, NTHR, 0, stream>>>(bias, ws, Bn);
  kapply<<<1024, NTHR, 0, stream>>>(x, ws, out, Bn);
}